// Policy_28656021799589
// MI455X (gfx1250) — compile-verified
//
#include <hip/hip_runtime.h>

typedef float v2f __attribute__((ext_vector_type(2)));
typedef float v8f __attribute__((ext_vector_type(8)));

#define B_DIM 128
#define P_DIM 128
#define D_DIM 128
#define N_DIM 4096
#define ROWS  16
#define PADN  (N_DIM + 4)          // +4 floats: lane-halves land 32 banks apart
#define CLIP  10.0f
#define SCALE 0.08838834764831845f // 1/sqrt(128)

__launch_bounds__(256, 1)
__global__ void fused_attn_probs(const float* __restrict__ A,
                                 const float* __restrict__ Km,
                                 const float* __restrict__ Mask,
                                 float* __restrict__ Out) {
  __shared__ alignas(16) float scores[ROWS][PADN];   // ~256 KB of the 320 KB WGP LDS
  __shared__ float rowsum[ROWS];
  __shared__ float rowinv[ROWS];

  const int tid  = threadIdx.x;
  const int lane = tid & 31;          // wave32
  const int wave = tid >> 5;
  const int hi   = lane >> 4;         // lane half (0: lanes 0-15, 1: lanes 16-31)
  const int lm   = lane & 15;

  const int b  = blockIdx.x >> 3;
  const int p0 = (blockIdx.x & 7) * ROWS;

  const int koff0 = hi * 2;           // A/B 32-bit layout: half 0 -> K%4 in {0,1}, half 1 -> {2,3}
  const int koff1 = koff0 + 1;

  // ---- stage A tile (16 x 128) into LDS coalesced (reuse scores storage) ----
  const float* Ablk = A + ((size_t)(b * P_DIM + p0)) * D_DIM;
  float* stage = &scores[0][0];
  for (int t = tid; t < ROWS * D_DIM; t += 256) stage[t] = Ablk[t];
  if (tid < ROWS) rowsum[tid] = 0.0f;
  __syncthreads();

  // per-lane A operands for all 32 K-steps (row m = lane%16), kept in VGPRs
  v2f areg[32];
  #pragma unroll
  for (int i = 0; i < 32; ++i) {
    areg[i].x = stage[lm * D_DIM + 4 * i + koff0];
    areg[i].y = stage[lm * D_DIM + 4 * i + koff1];
  }
  __syncthreads();   // scores LDS now free for exp values

  const float* Kb = Km   + (size_t)b * D_DIM * N_DIM;
  const float* Mb = Mask + ((size_t)(b * P_DIM + p0)) * N_DIM;

  float rsum[8];
  #pragma unroll
  for (int r = 0; r < 8; ++r) rsum[r] = 0.0f;

  // 256 column tiles of 16; wave w handles tiles w, w+8, ... (32 tiles)
  for (int tile = wave; tile < N_DIM / 16; tile += 8) {
    const int n0 = tile * 16;
    v8f acc = {0.f, 0.f, 0.f, 0.f, 0.f, 0.f, 0.f, 0.f};
    const float* kp0 = Kb + (size_t)koff0 * N_DIM + n0 + lm;
    const float* kp1 = Kb + (size_t)koff1 * N_DIM + n0 + lm;
    #pragma unroll
    for (int i = 0; i < 32; ++i) {     // D = 128 = 32 steps of K=4
      v2f bb;
      bb.x = kp0[(size_t)(4 * i) * N_DIM];
      bb.y = kp1[(size_t)(4 * i) * N_DIM];
      acc = __builtin_amdgcn_wmma_f32_16x16x4_f32(
          false, areg[i], false, bb, (short)0, acc, false, false);
    }
    // epilogue: scale, 10*tanh, +mask, exp(x - 10) (valid since clipped score <= 10)
    #pragma unroll
    for (int r = 0; r < 8; ++r) {
      const int m   = r + 8 * hi;      // C/D layout: VGPR r, lane half selects M or M+8
      const int col = n0 + lm;
      float s = acc[r] * SCALE;
#if __has_builtin(__builtin_amdgcn_tanhf)
      float th = __builtin_amdgcn_tanhf(s);
#else
      float e2 = __expf(2.0f * s);
      float th = 1.0f - 2.0f / (e2 + 1.0f);
#endif
      float v = CLIP * th + Mb[(size_t)m * N_DIM + col];
      float e = __expf(v - CLIP);
      scores[m][col] = e;
      rsum[r] += e;
    }
  }

  #pragma unroll
  for (int r = 0; r < 8; ++r)
    atomicAdd(&rowsum[r + 8 * hi], rsum[r]);   // ds_add_f32
  __syncthreads();
  if (tid < ROWS) rowinv[tid] = 1.0f / rowsum[tid];
  __syncthreads();

  // normalize + coalesced float4 stores
  float* Ob = Out + ((size_t)(b * P_DIM + p0)) * N_DIM;
  for (int j = tid; j < ROWS * (N_DIM / 4); j += 256) {
    const int row = j >> 10;           // 1024 float4 per row
    const int c4  = j & 1023;
    const float4 v  = *reinterpret_cast<const float4*>(&scores[row][c4 * 4]);
    const float inv = rowinv[row];
    float4 o = make_float4(v.x * inv, v.y * inv, v.z * inv, v.w * inv);
    *reinterpret_cast<float4*>(Ob + (size_t)row * N_DIM + c4 * 4) = o;
  }
}

extern "C" void kernel_launch(void* const* d_in, const int* in_sizes, int n_in,
                              void* d_out, int out_size, void* d_ws, size_t ws_size,
                              hipStream_t stream) {
  (void)in_sizes; (void)n_in; (void)out_size; (void)d_ws; (void)ws_size;
  const float* A    = (const float*)d_in[0];  // [B,P,D]
  const float* Kmat = (const float*)d_in[1];  // [B,D,N]
  const float* Mask = (const float*)d_in[2];  // [B,P,N]
  float* out = (float*)d_out;                 // [B,P,N]
  dim3 grid(B_DIM * (P_DIM / ROWS));          // 1024 workgroups
  fused_attn_probs<<<grid, dim3(256), 0, stream>>>(A, Kmat, Mask, out);
}